// Pointnet_SA_9792525435022
// MI455X (gfx1250) — compile-verified
//
#include <hip/hip_runtime.h>
#include <hip/hip_bf16.h>

// ---------------- problem constants ----------------
#define BATCH   8
#define NPTS    8192
#define NPOINT  1024
#define NSAMPLE 32
#define CIN     64
#define K0      68          // 67 channels padded to 68 (multiple of 4 for f32 WMMA)
#define F0      64
#define F1      64
#define F2      128
#define MROWS   (BATCH * NPOINT * NSAMPLE)   // 262144
#define RADIUS2 0.04f
#define BN_EPS  1e-3f

typedef __attribute__((ext_vector_type(2))) float v2f;
typedef __attribute__((ext_vector_type(8))) float v8f;

// =====================================================================
// 1. Farthest point sampling: one block per batch, 1024 threads,
//    each thread owns 8 points in registers. Argmax reduction carries
//    the winner's coordinates so the next centroid needs no global load.
// =====================================================================
__global__ __launch_bounds__(1024) void fps_kernel(const float* __restrict__ xyz,
                                                   float* __restrict__ new_xyz)
{
    const int b    = blockIdx.x;
    const int tid  = threadIdx.x;
    const int wave = tid >> 5, lane = tid & 31;
    const float* xb = xyz + (size_t)b * NPTS * 3;

    float px[8], py[8], pz[8], dist[8];
#pragma unroll
    for (int i = 0; i < 8; ++i) {
        int p = tid + (i << 10);
        px[i] = xb[p * 3 + 0];
        py[i] = xb[p * 3 + 1];
        pz[i] = xb[p * 3 + 2];
        dist[i] = 1e10f;
    }

    __shared__ float s_cx, s_cy, s_cz;
    __shared__ float rv[32], rx[32], ry[32], rz[32];
    __shared__ int   ri[32];

    if (tid == 0) {
        s_cx = px[0]; s_cy = py[0]; s_cz = pz[0];
        float* o = new_xyz + (size_t)b * NPOINT * 3;
        o[0] = px[0]; o[1] = py[0]; o[2] = pz[0];
    }
    __syncthreads();

    for (int s = 1; s < NPOINT; ++s) {
        const float cx = s_cx, cy = s_cy, cz = s_cz;
        float bv = -1.0f, bx = 0.f, by = 0.f, bz = 0.f;
        int   bi = 0x7fffffff;
#pragma unroll
        for (int i = 0; i < 8; ++i) {
            float dx = px[i] - cx, dy = py[i] - cy, dz = pz[i] - cz;
            float d  = dx * dx + dy * dy + dz * dz;
            d = fminf(dist[i], d);
            dist[i] = d;
            int p = tid + (i << 10);
            if (d > bv || (d == bv && p < bi)) { bv = d; bi = p; bx = px[i]; by = py[i]; bz = pz[i]; }
        }
        // wave32 reduction (max value, ties -> lowest index), coords ride along
#pragma unroll
        for (int off = 16; off > 0; off >>= 1) {
            float ov = __shfl_xor(bv, off, 32);
            int   oi = __shfl_xor(bi, off, 32);
            float ox = __shfl_xor(bx, off, 32);
            float oy = __shfl_xor(by, off, 32);
            float oz = __shfl_xor(bz, off, 32);
            if (ov > bv || (ov == bv && oi < bi)) { bv = ov; bi = oi; bx = ox; by = oy; bz = oz; }
        }
        if (lane == 0) { rv[wave] = bv; ri[wave] = bi; rx[wave] = bx; ry[wave] = by; rz[wave] = bz; }
        __syncthreads();
        if (wave == 0) {
            bv = rv[lane]; bi = ri[lane]; bx = rx[lane]; by = ry[lane]; bz = rz[lane];
#pragma unroll
            for (int off = 16; off > 0; off >>= 1) {
                float ov = __shfl_xor(bv, off, 32);
                int   oi = __shfl_xor(bi, off, 32);
                float ox = __shfl_xor(bx, off, 32);
                float oy = __shfl_xor(by, off, 32);
                float oz = __shfl_xor(bz, off, 32);
                if (ov > bv || (ov == bv && oi < bi)) { bv = ov; bi = oi; bx = ox; by = oy; bz = oz; }
            }
            if (lane == 0) {
                s_cx = bx; s_cy = by; s_cz = bz;
                float* o = new_xyz + ((size_t)b * NPOINT + s) * 3;
                o[0] = bx; o[1] = by; o[2] = bz;
            }
        }
        __syncthreads();
    }
}

// =====================================================================
// 2. Ball query: one wave per centroid. Equivalent to the reference's
//    sort-by-index + take-first-32 + pad-with-first.
// =====================================================================
__global__ __launch_bounds__(256) void ballquery_kernel(const float* __restrict__ xyz,
                                                        const float* __restrict__ new_xyz,
                                                        int* __restrict__ ball_idx)
{
    const int wave = threadIdx.x >> 5, lane = threadIdx.x & 31;
    const int g = blockIdx.x * 8 + wave;           // centroid id in [0, B*S)
    const int b = g >> 10;                         // / NPOINT
    const float* xb = xyz + (size_t)b * NPTS * 3;
    const float cx = new_xyz[g * 3 + 0];
    const float cy = new_xyz[g * 3 + 1];
    const float cz = new_xyz[g * 3 + 2];
    int* out = ball_idx + (size_t)g * NSAMPLE;

    int cnt = 0;
    int first_idx = -1;
    for (int base = 0; base < NPTS && cnt < NSAMPLE; base += 32) {
        int p = base + lane;
        float dx = xb[p * 3 + 0] - cx;
        float dy = xb[p * 3 + 1] - cy;
        float dz = xb[p * 3 + 2] - cz;
        bool within = (dx * dx + dy * dy + dz * dz) <= RADIUS2;
        unsigned mask = (unsigned)__ballot(within);
        if (first_idx < 0 && mask) first_idx = base + __builtin_ctz(mask);
        int pos = cnt + __builtin_popcount(mask & ((1u << lane) - 1u));
        if (within && pos < NSAMPLE) out[pos] = p;
        cnt += __builtin_popcount(mask);
    }
    if (first_idx < 0) first_idx = 0;              // cannot happen (centroid is in its own ball)
    for (int k = cnt + lane; k < NSAMPLE; k += 32) out[k] = first_idx;
}

// =====================================================================
// 3. Gather + concat into x0: (MROWS, 68) fp32 row-major.
//    cols 0-2: grouped_xyz - centroid, cols 3-66: points, col 67: 0 pad.
// =====================================================================
__global__ __launch_bounds__(256) void gather_kernel(const float* __restrict__ xyz,
                                                     const float* __restrict__ points,
                                                     const float* __restrict__ new_xyz,
                                                     const int* __restrict__ ball_idx,
                                                     float* __restrict__ x0)
{
    int t   = blockIdx.x * 256 + threadIdx.x;      // < MROWS * K0
    int row = t / K0;
    int c   = t - row * K0;
    int b   = row >> 15;                           // / (NPOINT*NSAMPLE)
    int rem = row & 32767;
    int s   = rem >> 5;
    int idx = ball_idx[row];
    float v;
    if (c < 3)
        v = xyz[((size_t)b * NPTS + idx) * 3 + c] - new_xyz[((size_t)b * NPOINT + s) * 3 + c];
    else if (c < 67)
        v = points[((size_t)b * NPTS + idx) * CIN + (c - 3)];
    else
        v = 0.f;
    x0[t] = v;
}

// pad W0 (67x64) -> (68x64) with zero row
__global__ void w0pad_kernel(const float* __restrict__ w0, float* __restrict__ w0p)
{
    int t = blockIdx.x * 256 + threadIdx.x;
    if (t < K0 * F0) {
        int r = t / F0;
        w0p[t] = (r < 67) ? w0[t] : 0.f;
    }
}

__global__ void zero_kernel(float* __restrict__ p, int n)
{
    int t = blockIdx.x * 256 + threadIdx.x;
    if (t < n) p[t] = 0.f;
}

// =====================================================================
// 4. GEMM (fp32 WMMA 16x16x4) + fused input BN/ReLU + BN stats output.
//    One wave computes a 16-row stripe x all FOUT columns (A-frag reuse).
//    A frag per lane: rows row0+lane%16, cols ka..ka+1 (b64 load)
//    B frag per lane: W[ka][n], W[ka+1][n]   n = 16*t + lane%16
//    C/D layout: vgpr v -> row v + 8*(lane/16), col lane%16.
// =====================================================================
template <int KDIM, int FOUT, bool RELUIN>
__global__ __launch_bounds__(256) void gemm_bn_kernel(const float* __restrict__ A,
                                                      const float* __restrict__ W,
                                                      const float* __restrict__ coef,
                                                      float* __restrict__ Y,
                                                      float* __restrict__ accum)
{
    constexpr int CT = FOUT / 16;
    const int wave = threadIdx.x >> 5, lane = threadIdx.x & 31;
    const int half = lane >> 4, lm = lane & 15;
    const int row0 = (blockIdx.x * 8 + wave) << 4;
    const float* Arow = A + (size_t)(row0 + lm) * KDIM;

    const v8f vzero = {0.f, 0.f, 0.f, 0.f, 0.f, 0.f, 0.f, 0.f};
    v8f c[CT];
#pragma unroll
    for (int t = 0; t < CT; ++t) c[t] = vzero;

#pragma unroll
    for (int kk = 0; kk < KDIM; kk += 4) {
        const int ka = kk + 2 * half;
        v2f a = *(const v2f*)(Arow + ka);
        if constexpr (RELUIN) {   // fused BN+ReLU of previous layer on A load
            a.x = fmaxf(fmaf(a.x, coef[ka],     coef[KDIM + ka]),     0.f);
            a.y = fmaxf(fmaf(a.y, coef[ka + 1], coef[KDIM + ka + 1]), 0.f);
        }
#pragma unroll
        for (int t = 0; t < CT; ++t) {
            v2f bf;
            bf.x = W[(size_t)ka * FOUT + t * 16 + lm];
            bf.y = W[(size_t)(ka + 1) * FOUT + t * 16 + lm];
            c[t] = __builtin_amdgcn_wmma_f32_16x16x4_f32(
                false, a, false, bf, (short)0, c[t], false, false);
        }
    }

    __shared__ float ls[8][FOUT], lq[8][FOUT];
#pragma unroll
    for (int t = 0; t < CT; ++t) {
        float s = 0.f, q = 0.f;
#pragma unroll
        for (int v = 0; v < 8; ++v) {
            float val = c[t][v];
            Y[(size_t)(row0 + v + 8 * half) * FOUT + t * 16 + lm] = val;
            s += val;
            q += val * val;
        }
        s += __shfl_down(s, 16, 32);
        q += __shfl_down(q, 16, 32);
        if (half == 0) { ls[wave][t * 16 + lm] = s; lq[wave][t * 16 + lm] = q; }
    }
    __syncthreads();
    if (threadIdx.x < FOUT) {
        float ts = 0.f, tq = 0.f;
#pragma unroll
        for (int wv = 0; wv < 8; ++wv) { ts += ls[wv][threadIdx.x]; tq += lq[wv][threadIdx.x]; }
        atomicAdd(&accum[threadIdx.x], ts);
        atomicAdd(&accum[FOUT + threadIdx.x], tq);
    }
}

// fold (gamma, beta, mean, var) into per-channel scale/shift
__global__ void bn_finalize_kernel(const float* __restrict__ accum,
                                   const float* __restrict__ gamma,
                                   const float* __restrict__ beta,
                                   float* __restrict__ coef, int F, float invM)
{
    int f = threadIdx.x;
    if (f < F) {
        float mean = accum[f] * invM;
        float var  = accum[F + f] * invM - mean * mean;
        float sc   = gamma[f] * rsqrtf(var + BN_EPS);
        coef[f]     = sc;
        coef[F + f] = beta[f] - mean * sc;
    }
}

// =====================================================================
// 5. Fused final BN+ReLU + max over the K=32 samples -> new_points
// =====================================================================
__global__ __launch_bounds__(256) void maxpool_kernel(const float* __restrict__ Y,
                                                      const float* __restrict__ coef,
                                                      float* __restrict__ out)
{
    int t  = blockIdx.x * 256 + threadIdx.x;   // < B*S*F2
    int f  = t & (F2 - 1);
    int bs = t >> 7;
    float sc = coef[f], sh = coef[F2 + f];
    const float* y = Y + (size_t)bs * NSAMPLE * F2 + f;
    float m = 0.f;                             // relu floor
#pragma unroll 4
    for (int k = 0; k < NSAMPLE; ++k)
        m = fmaxf(m, fmaf(y[(size_t)k * F2], sc, sh));
    out[t] = m;                                // == max_k relu(sc*y+sh)
}

// =====================================================================
// launch
// =====================================================================
extern "C" void kernel_launch(void* const* d_in, const int* in_sizes, int n_in,
                              void* d_out, int out_size, void* d_ws, size_t ws_size,
                              hipStream_t stream)
{
    (void)in_sizes; (void)n_in; (void)out_size; (void)ws_size;

    const float* xyz    = (const float*)d_in[0];
    const float* points = (const float*)d_in[1];
    const float* w0 = (const float*)d_in[2];
    const float* g0 = (const float*)d_in[3];
    const float* b0 = (const float*)d_in[4];
    const float* w1 = (const float*)d_in[5];
    const float* g1 = (const float*)d_in[6];
    const float* b1 = (const float*)d_in[7];
    const float* w2 = (const float*)d_in[8];
    const float* g2 = (const float*)d_in[9];
    const float* b2 = (const float*)d_in[10];

    float* out_xyz = (float*)d_out;                             // (8,1024,3)
    float* out_pts = (float*)d_out + BATCH * NPOINT * 3;        // (8,1024,128)

    // ---- workspace layout (~208 MB) ----
    char* ws = (char*)d_ws;
    int*   ball = (int*)ws;                                     // 1 MB
    float* w0p  = (float*)(ws + (1u << 20));                    // 17408 B
    float* bn   = (float*)(ws + (1u << 20) + 32768);            // 768 accum floats
    float* coef = bn + 768;                                     // 768 coef floats
    float* x0   = (float*)(ws + 2 * (size_t)(1u << 20));        // MROWS*68 fp32  (71.3 MB)
    float* y0   = (float*)(ws + 2 * (size_t)(1u << 20)
                              + (size_t)MROWS * K0 * sizeof(float)); // MROWS*64 / y2 base
    float* y1 = x0;   // layer-1 output reuses x0's space (x0 dead after gemm0)
    float* y2 = y0;   // layer-2 output reuses y0's space + tail (y0 dead after gemm1 input read... read fully before? gemm2 reads y1, writes y2)
    float* bn0 = bn,        *bn1 = bn + 256,   *bn2 = bn + 512;
    float* cf0 = coef,      *cf1 = coef + 256, *cf2 = coef + 512;

    const float invM = 1.0f / (float)MROWS;

    fps_kernel<<<BATCH, 1024, 0, stream>>>(xyz, out_xyz);
    ballquery_kernel<<<(BATCH * NPOINT) / 8, 256, 0, stream>>>(xyz, out_xyz, ball);
    gather_kernel<<<(MROWS * K0) / 256, 256, 0, stream>>>(xyz, points, out_xyz, ball, x0);
    w0pad_kernel<<<(K0 * F0 + 255) / 256, 256, 0, stream>>>(w0, w0p);
    zero_kernel<<<3, 256, 0, stream>>>(bn, 768);

    const int rowBlocks = MROWS / (16 * 8);   // 2048 blocks, 8 waves x 16 rows each

    gemm_bn_kernel<K0, F0, false><<<rowBlocks, 256, 0, stream>>>(x0, w0p, nullptr, y0, bn0);
    bn_finalize_kernel<<<1, 128, 0, stream>>>(bn0, g0, b0, cf0, F0, invM);

    gemm_bn_kernel<64, F1, true><<<rowBlocks, 256, 0, stream>>>(y0, w1, cf0, y1, bn1);
    bn_finalize_kernel<<<1, 128, 0, stream>>>(bn1, g1, b1, cf1, F1, invM);

    gemm_bn_kernel<64, F2, true><<<rowBlocks, 256, 0, stream>>>(y1, w2, cf1, y2, bn2);
    bn_finalize_kernel<<<1, 128, 0, stream>>>(bn2, g2, b2, cf2, F2, invM);

    maxpool_kernel<<<(BATCH * NPOINT * F2) / 256, 256, 0, stream>>>(y2, cf2, out_pts);
}